// Warp_14877766714074
// MI455X (gfx1250) — compile-verified
//
#include <hip/hip_runtime.h>

// Bilinear flow warp, channels_last.
//   img: [8, 384, 512, 64] f32
//   flo: [8, 384, 512, 2]  f32
//   out: [8, 384, 512, 64] f32
//
// Bandwidth-bound gather (~1 FLOP/byte, ~820 MB min traffic -> ~35us floor
// at 23.3 TB/s). 16 lanes per pixel, one 4-channel group per lane ->
// 256B coalesced b128 bursts per corner gather; NT b128 stores keep L2
// resident with img. Grid encodes (b, y, x-tile) so there is no div/mod
// index recovery in the kernel.

#define BB 8
#define HH 384
#define WW 512
#define CC 64

// Native Clang vector types (required by __builtin_nontemporal_*)
typedef __attribute__((ext_vector_type(2))) float v2f;
typedef __attribute__((ext_vector_type(4))) float v4f;

__global__ __launch_bounds__(256) void warp_bilinear_kernel(
    const float* __restrict__ img,
    const float* __restrict__ flo,
    float* __restrict__ out)
{
    // grid: (W/16, H, B); block: 256 threads = 16 pixels of one row
    const unsigned x  = (blockIdx.x << 4) + (threadIdx.x >> 4); // pixel x
    const unsigned y  = blockIdx.y;                             // pixel y (uniform)
    const unsigned b  = blockIdx.z;                             // batch   (uniform)
    const unsigned cw = threadIdx.x & 15u;                      // float4 group

    const unsigned row = (b * HH + y);          // block-uniform
    const unsigned p   = (row << 9) + x;        // pixel id (W = 512)

    // Flow (dx, dy) — one-shot streaming read
    const v2f f = __builtin_nontemporal_load((const v2f*)flo + p);

    const float sx  = (float)x + f.x;
    const float sy  = (float)y + f.y;
    const float x0f = floorf(sx);
    const float y0f = floorf(sy);
    const float wx  = sx - x0f;                 // in [0,1)
    const float wy  = sy - y0f;

    const int ix0 = (int)x0f;                   // exact: floorf output is integral
    const int iy0 = (int)y0f;
    const int x0 = min(max(ix0,     0), WW - 1);  // v_med3
    const int x1 = min(max(ix0 + 1, 0), WW - 1);
    const int y0 = min(max(iy0,     0), HH - 1);
    const int y1 = min(max(iy0 + 1, 0), HH - 1);

    // float4 indices: shifts only (W=512 -> <<9, 16 float4/pixel -> <<4)
    const v4f* __restrict__ img4 = (const v4f*)img;
    const unsigned base = b * (unsigned)(HH * WW);              // uniform
    const unsigned ra0  = base + ((unsigned)y0 << 9);
    const unsigned ra1  = base + ((unsigned)y1 << 9);

    const v4f Ia = img4[((ra0 + (unsigned)x0) << 4) + cw];      // top-left
    const v4f Ib = img4[((ra1 + (unsigned)x0) << 4) + cw];      // bottom-left
    const v4f Ic = img4[((ra0 + (unsigned)x1) << 4) + cw];      // top-right
    const v4f Id = img4[((ra1 + (unsigned)x1) << 4) + cw];      // bottom-right

    const float wa = (1.0f - wx) * (1.0f - wy);
    const float wb = (1.0f - wx) * wy;
    const float wc = wx * (1.0f - wy);
    const float wd = wx * wy;

    v4f r;
    r.x = fmaf(wd, Id.x, fmaf(wc, Ic.x, fmaf(wb, Ib.x, wa * Ia.x)));
    r.y = fmaf(wd, Id.y, fmaf(wc, Ic.y, fmaf(wb, Ib.y, wa * Ia.y)));
    r.z = fmaf(wd, Id.z, fmaf(wc, Ic.z, fmaf(wb, Ib.z, wa * Ia.z)));
    r.w = fmaf(wd, Id.w, fmaf(wc, Ic.w, fmaf(wb, Ib.w, wa * Ia.w)));

    // Streamed output: non-temporal b128 store
    __builtin_nontemporal_store(r, (v4f*)out + (p << 4) + cw);
}

extern "C" void kernel_launch(void* const* d_in, const int* in_sizes, int n_in,
                              void* d_out, int out_size, void* d_ws, size_t ws_size,
                              hipStream_t stream) {
    const float* img = (const float*)d_in[0];
    const float* flo = (const float*)d_in[1];
    float* out       = (float*)d_out;

    // 16 pixels per 256-thread block (8 wave32); dims divide exactly.
    dim3 grid(WW / 16, HH, BB);   // (32, 384, 8) = 98,304 blocks
    dim3 block(256);
    warp_bilinear_kernel<<<grid, block, 0, stream>>>(img, flo, out);
}